// VQ_EMA_91087666413894
// MI455X (gfx1250) — compile-verified
//
#include <hip/hip_runtime.h>
#include <hip/hip_bf16.h>

#define LATENT_DIM 64
#define CODEBOOK_SIZE 8192
#define N_TOKENS 32768
#define GAMMA 0.99f
#define COMMITMENT_COST 1.0f
#define EPSILON 1e-9f

typedef __attribute__((ext_vector_type(2))) float v2f;
typedef __attribute__((ext_vector_type(8))) float v8f;

// ---------------------------------------------------------------------------
// Kernel 0: zero workspace accumulators (cluster_size, dw, loss)
// ---------------------------------------------------------------------------
__global__ void vq_zero_kernel(float* __restrict__ p, int n) {
    int i = blockIdx.x * blockDim.x + threadIdx.x;
    if (i < n) p[i] = 0.0f;
}

// ---------------------------------------------------------------------------
// Kernel 1: cnorm[k] = sum_d codebook[d,k]^2
// ---------------------------------------------------------------------------
__global__ void vq_cnorm_kernel(const float* __restrict__ cb, float* __restrict__ cnorm) {
    int k = blockIdx.x * blockDim.x + threadIdx.x;
    if (k >= CODEBOOK_SIZE) return;
    float s = 0.0f;
    #pragma unroll 8
    for (int d = 0; d < LATENT_DIM; ++d) {
        float v = cb[d * CODEBOOK_SIZE + k];
        s += v * v;
    }
    cnorm[k] = s;
}

// ---------------------------------------------------------------------------
// CDNA5 async global->LDS copy (128b per lane), tracked with ASYNCcnt.
// lds_off is the 32-bit LDS byte offset (low 32 bits of a local->flat cast).
// ---------------------------------------------------------------------------
__device__ __forceinline__ void async_stage_b128(uint32_t lds_off, const float* gptr) {
    asm volatile("global_load_async_to_lds_b128 %0, %1, off"
                 :: "v"(lds_off), "v"(gptr) : "memory");
}
__device__ __forceinline__ void wait_async0() {
    asm volatile("s_wait_asynccnt 0x0" ::: "memory");
}

// ---------------------------------------------------------------------------
// Kernel 2: fused GEMM (-2 z@C + ||c||^2) + argmin via f32 WMMA.
// Workgroup = 256 threads = 8 waves; wave w owns 16 token rows. A 64x16
// codebook column tile is async-DMA'd into a ping-pong LDS buffer (one
// barrier per tile, next tile's fetch overlapped with current WMMA chain).
// ---------------------------------------------------------------------------
__global__ __launch_bounds__(256) void vq_argmin_kernel(
    const float* __restrict__ z, const float* __restrict__ cb,
    const float* __restrict__ cnorm, int* __restrict__ idx_out,
    float* __restrict__ idxf_out)
{
    __shared__ __align__(16) float lds[2][64 * 20];  // pad stride 20 (bank-safe)

    const int lane = threadIdx.x & 31;
    const int wave = threadIdx.x >> 5;
    const int row0 = blockIdx.x * 128 + wave * 16;
    const int m    = lane & 15;         // row (A) / col (B, C/D) within tile
    const int hi   = lane >> 4;         // half-wave select
    const int koff = hi * 2;            // K sub-offset per ISA 16x4 f32 layout

    // this thread's staging slot: latent row r, column quad q
    const int r = threadIdx.x >> 2;     // 0..63
    const int q = threadIdx.x & 3;      // 0..3
    const float* gsrc = cb + (size_t)r * CODEBOOK_SIZE + q * 4;   // +col0 per tile
    const uint32_t ldsoff0 = (uint32_t)(uintptr_t)&lds[0][r * 20 + q * 4];
    const uint32_t ldsoff1 = (uint32_t)(uintptr_t)&lds[1][r * 20 + q * 4];

    // ---- Load A tile once: 16 rows x 64 latents, as 16 k-steps of v2f ----
    v2f a[16];
    const float* zrow = z + (size_t)(row0 + m) * LATENT_DIM;
    #pragma unroll
    for (int t = 0; t < 16; ++t) {
        a[t].x = zrow[4 * t + koff + 0];
        a[t].y = zrow[4 * t + koff + 1];
    }

    float minv[8];
    int   mini[8];
    #pragma unroll
    for (int i = 0; i < 8; ++i) { minv[i] = 3.4e38f; mini[i] = 0; }

    // prologue: async-stage tile 0 into buffer 0
    async_stage_b128(ldsoff0, gsrc);

    const int NTILES = CODEBOOK_SIZE / 16;
    for (int it = 0; it < NTILES; ++it) {
        const int cur  = it & 1;
        const int col0 = it * 16;

        wait_async0();        // my DMA into buf[cur] complete
        __syncthreads();      // all waves' DMAs visible; prior reads of buf[cur^1] done

        if (it + 1 < NTILES)  // prefetch next tile into the other buffer (overlapped)
            async_stage_b128(cur ? ldsoff0 : ldsoff1, gsrc + (it + 1) * 16);

        const float* tile = lds[cur];
        v8f acc = {};
        #pragma unroll
        for (int t = 0; t < 16; ++t) {
            v2f b;
            b.x = tile[(4 * t + koff + 0) * 20 + m];
            b.y = tile[(4 * t + koff + 1) * 20 + m];
            acc = __builtin_amdgcn_wmma_f32_16x16x4_f32(
                /*neg_a=*/false, a[t], /*neg_b=*/false, b,
                /*c_mod=*/(short)0, acc, /*reuse_a=*/false, /*reuse_b=*/false);
        }

        // epilogue: dist = ||c||^2 - 2 * (z . c); running argmin per row slot
        const float cn = cnorm[col0 + m];
        const int   ci = col0 + m;
        #pragma unroll
        for (int i = 0; i < 8; ++i) {
            float dist = cn - 2.0f * acc[i];
            if (dist < minv[i]) { minv[i] = dist; mini[i] = ci; }  // '<' keeps first tie
        }
    }

    // ---- cross-lane argmin within each 16-lane half (first-index ties) ----
    #pragma unroll
    for (int i = 0; i < 8; ++i) {
        float v  = minv[i];
        int   id = mini[i];
        #pragma unroll
        for (int s = 1; s < 16; s <<= 1) {
            float ov = __shfl_xor(v, s, 32);
            int   oi = __shfl_xor(id, s, 32);
            if (ov < v || (ov == v && oi < id)) { v = ov; id = oi; }
        }
        minv[i] = v; mini[i] = id;
    }

    if (m == 0) {
        #pragma unroll
        for (int i = 0; i < 8; ++i) {
            int rr = row0 + hi * 8 + i;  // lanes 0/16 hold rows 0-7 / 8-15 of tile
            idx_out[rr]  = mini[i];
            idxf_out[rr] = (float)mini[i];
        }
    }
}

// ---------------------------------------------------------------------------
// Kernel 3: gather quantised, write quantised_st, accumulate dw / cluster /
// loss. One thread per (token, dim) element.
// ---------------------------------------------------------------------------
__global__ __launch_bounds__(256) void vq_scatter_kernel(
    const float* __restrict__ z, const float* __restrict__ cb,
    const int* __restrict__ idx, float* __restrict__ q_out,
    float* __restrict__ dw, float* __restrict__ cs, float* __restrict__ loss)
{
    __shared__ float red[256];
    int t = blockIdx.x * 256 + threadIdx.x;        // over N_TOKENS*64
    int n = t >> 6;
    int d = t & 63;
    int k = idx[n];
    float zv = z[t];
    float qv = cb[d * CODEBOOK_SIZE + k];
    q_out[t] = qv;                                  // quantised_st forward value
    float diff = qv - zv;
    atomicAdd(&dw[d * CODEBOOK_SIZE + k], zv);      // dw[d,k] += z[n,d]
    if (d == 0) atomicAdd(&cs[k], 1.0f);

    red[threadIdx.x] = diff * diff;
    __syncthreads();
    #pragma unroll
    for (int s = 128; s > 0; s >>= 1) {
        if ((int)threadIdx.x < s) red[threadIdx.x] += red[threadIdx.x + s];
        __syncthreads();
    }
    if (threadIdx.x == 0) atomicAdd(loss, red[0]);
}

// ---------------------------------------------------------------------------
// Kernel 4: EMA codebook update, avg_probs, commitment loss.
// ---------------------------------------------------------------------------
__global__ void vq_finalize_kernel(
    const float* __restrict__ cb, const float* __restrict__ dw,
    const float* __restrict__ cs, const float* __restrict__ loss,
    float* __restrict__ ncb_out, float* __restrict__ avg_out,
    float* __restrict__ loss_out)
{
    int t = blockIdx.x * 256 + threadIdx.x;        // over 64*8192
    int d = t >> 13;
    int k = t & (CODEBOOK_SIZE - 1);
    float c = cs[k];
    ncb_out[t] = GAMMA * cb[t] + (1.0f - GAMMA) * dw[t] / (c + EPSILON);
    if (d == 0) avg_out[k] = c * (1.0f / (float)N_TOKENS);
    if (t == 0) loss_out[0] = COMMITMENT_COST * loss[0] *
                              (1.0f / (float)(N_TOKENS * LATENT_DIM));
}

// ---------------------------------------------------------------------------
extern "C" void kernel_launch(void* const* d_in, const int* in_sizes, int n_in,
                              void* d_out, int out_size, void* d_ws, size_t ws_size,
                              hipStream_t stream) {
    const float* z  = (const float*)d_in[0];   // [32768, 64]
    const float* cb = (const float*)d_in[1];   // [64, 8192]

    float* out      = (float*)d_out;
    float* q_out    = out;                       // 2097152
    float* loss_out = out + 2097152;             // 1
    float* avg_out  = out + 2097153;             // 8192
    float* idxf_out = out + 2105345;             // 32768
    float* ncb_out  = out + 2138113;             // 524288

    float* ws    = (float*)d_ws;
    float* cnorm = ws;                           // 8192
    float* cs    = ws + 8192;                    // 8192
    float* dw    = ws + 16384;                   // 524288
    float* loss  = ws + 540672;                  // 1
    int*   idx   = (int*)(ws + 540673);          // 32768 ints

    // zero cs + dw + loss (contiguous: 8192 + 524288 + 1 floats)
    {
        int n = 8192 + 64 * CODEBOOK_SIZE + 1;
        vq_zero_kernel<<<(n + 255) / 256, 256, 0, stream>>>(cs, n);
    }
    vq_cnorm_kernel<<<CODEBOOK_SIZE / 256, 256, 0, stream>>>(cb, cnorm);
    vq_argmin_kernel<<<N_TOKENS / 128, 256, 0, stream>>>(z, cb, cnorm, idx, idxf_out);
    vq_scatter_kernel<<<(N_TOKENS * LATENT_DIM) / 256, 256, 0, stream>>>(
        z, cb, idx, q_out, dw, cs, loss);
    vq_finalize_kernel<<<(LATENT_DIM * CODEBOOK_SIZE) / 256, 256, 0, stream>>>(
        cb, dw, cs, loss, ncb_out, avg_out, loss_out);
}